// RotationDeltaLoss_16415365005543
// MI455X (gfx1250) — compile-verified
//
#include <hip/hip_runtime.h>
#include <stdint.h>

// ---------------------------------------------------------------------------
// RotationDeltaLoss on MI455X (gfx1250).
// Memory-bound streaming reduction: 302 MB in / 4 B out -> ~13 us @ 23.3 TB/s.
// CDNA5 path: async global->LDS staging (global_load_async_to_lds_b128,
// s_wait_asynccnt, double-buffered) so HBM traffic is 128-bit fully coalesced
// while the 36-byte-strided row reads are served by LDS (stride 9 dwords is
// bank-conflict-free across 64 banks). Wave32 shfl reductions, two-pass
// deterministic final reduction (no float atomics).
// ---------------------------------------------------------------------------

#define EPSF 1e-8f
#define TILE_ROWS   256
#define TILE_FLOATS (TILE_ROWS * 9)      // 2304 floats per array per tile
#define BUF_FLOATS  (2 * TILE_FLOATS)    // pred + target = 4608 floats
#define NCHUNK      1152                 // 16B chunks per tile (both arrays)
#define CPT         5                    // chunks per thread (uniform, w/ wrap)

#define WAIT_ASYNC(N) asm volatile("s_wait_asynccnt %0" :: "n"(N) : "memory")

struct F3 { float x, y, z; };

__device__ __forceinline__ float dot3(F3 a, F3 b) {
    return fmaf(a.x, b.x, fmaf(a.y, b.y, a.z * b.z));
}
__device__ __forceinline__ F3 sc3(F3 a, float s) { return {a.x*s, a.y*s, a.z*s}; }
__device__ __forceinline__ F3 sub3(F3 a, F3 b)   { return {a.x-b.x, a.y-b.y, a.z-b.z}; }
__device__ __forceinline__ F3 fnm3(F3 a, float s, F3 e) {
    return {fmaf(-s, e.x, a.x), fmaf(-s, e.y, a.y), fmaf(-s, e.z, a.z)};
}
__device__ __forceinline__ F3 cross3(F3 a, F3 b) {
    return {a.y*b.z - a.z*b.y, a.z*b.x - a.x*b.z, a.x*b.y - a.y*b.x};
}
__device__ __forceinline__ float clip10f(float v) {
    return fminf(fmaxf(v, -10.0f), 10.0f);
}

__device__ __forceinline__ void gram_schmidt(F3 a1, F3 a2, F3 a3,
                                             F3& e1, F3& e2, F3& e3)
{
    e1 = sc3(a1, 1.0f / (sqrtf(dot3(a1, a1)) + EPSF));
    F3 u2 = fnm3(a2, dot3(e1, a2), e1);
    e2 = sc3(u2, 1.0f / (sqrtf(dot3(u2, u2)) + EPSF));
    F3 u3 = fnm3(fnm3(a3, dot3(e1, a3), e1), dot3(e2, a3), e2);
    e3 = sc3(u3, 1.0f / (sqrtf(dot3(u3, u3)) + EPSF));
    float det = dot3(e1, cross3(e2, e3));       // det([e1 e2 e3]) = e1.(e2 x e3)
    if (det < 0.0f) { e3.x = -e3.x; e3.y = -e3.y; e3.z = -e3.z; }
}

// Issue the async copies for one 256-row tile (pred + target) into an LDS
// buffer. 1152 16-byte chunks, 5 per thread (the last 128 wrap and duplicate
// chunks 0..127 with identical data -> harmless, keeps the per-wave ASYNCcnt
// increment uniform so a constant s_wait_asynccnt works for double buffering).
__device__ __forceinline__ void issue_tile(const float* __restrict__ pred,
                                           const float* __restrict__ targ,
                                           float* ldsBase,
                                           unsigned tileByteBase,
                                           unsigned maxByte)
{
#pragma unroll
    for (int k = 0; k < CPT; ++k) {
        int c = (int)threadIdx.x + k * 256;
        if (c >= NCHUNK) c -= NCHUNK;                 // wrap duplicates
        int isT = (c >= NCHUNK / 2);
        int cc  = isT ? (c - NCHUNK / 2) : c;         // chunk within array
        unsigned gOff = tileByteBase + (unsigned)cc * 16u;
        if (gOff > maxByte) gOff = maxByte;           // clamp OOB tail chunks
        unsigned long long gbase =
            (unsigned long long)(uintptr_t)(isT ? targ : pred);
        float* dst = ldsBase + (isT ? TILE_FLOATS : 0) + cc * 4;
        unsigned ldsAddr = (unsigned)(uintptr_t)dst;  // low 32 bits = LDS offset
        asm volatile("global_load_async_to_lds_b128 %0, %1, %2"
                     :: "v"(ldsAddr), "v"(gOff), "s"(gbase)
                     : "memory");
    }
}

__global__ __launch_bounds__(256) void rot_loss_main(
    const float* __restrict__ pred, const float* __restrict__ targ,
    float* __restrict__ partials, int nTiles, long long nRows)
{
    __shared__ __align__(16) float smem[2 * BUF_FLOATS];   // 36,864 B double buffer
    __shared__ float red[8 * 5];

    const unsigned maxByte =
        (unsigned)((((unsigned long long)nRows * 36ull) - 16ull) & ~15ull);

    float accCho = 0.f, accAng = 0.f, accOrt = 0.f, accL2 = 0.f, accFb = 0.f;

    int cur = 0;
    int t = (int)blockIdx.x;                 // grid-stride over tiles
    issue_tile(pred, targ, smem, (unsigned)t * (unsigned)(TILE_ROWS * 36), maxByte);

    while (t < nTiles) {
        int tn = t + (int)gridDim.x;
        if (tn < nTiles) {
            issue_tile(pred, targ, smem + (cur ^ 1) * BUF_FLOATS,
                       (unsigned)tn * (unsigned)(TILE_ROWS * 36), maxByte);
            WAIT_ASYNC(CPT);                 // retire previous tile's 5 copies
        } else {
            WAIT_ASYNC(0);
        }
        __syncthreads();                     // tile t fully visible in LDS

        {
            const float* lp = smem + cur * BUF_FLOATS + (int)threadIdx.x * 9;
            const float* lt = lp + TILE_FLOATS;
            long long r = (long long)t * TILE_ROWS + (long long)threadIdx.x;
            if (r < nRows) {
                float p[9], tg[9];
#pragma unroll
                for (int i = 0; i < 9; ++i) { p[i] = lp[i]; tg[i] = lt[i]; }
                float cp[9], ct[9];
#pragma unroll
                for (int i = 0; i < 9; ++i) { cp[i] = clip10f(p[i]); ct[i] = clip10f(tg[i]); }

                // columns of the 3x3 (row-major storage: elem(i,j) = v[3i+j])
                F3 pa1 = {cp[0], cp[3], cp[6]};
                F3 pa2 = {cp[1], cp[4], cp[7]};
                F3 pa3 = {cp[2], cp[5], cp[8]};
                F3 ta1 = {ct[0], ct[3], ct[6]};
                F3 ta2 = {ct[1], ct[4], ct[7]};
                F3 ta3 = {ct[2], ct[5], ct[8]};

                F3 pe1, pe2, pe3, te1, te2, te3;
                gram_schmidt(pa1, pa2, pa3, pe1, pe2, pe3);
                gram_schmidt(ta1, ta2, ta3, te1, te2, te3);

                // chordal: ||P - T||_F^2
                F3 d1 = sub3(pe1, te1), d2 = sub3(pe2, te2), d3 = sub3(pe3, te3);
                accCho += dot3(d1, d1) + dot3(d2, d2) + dot3(d3, d3);

                // angular: trace(P^T T) via column dots, clip, acos
                float trc = dot3(pe1, te1) + dot3(pe2, te2) + dot3(pe3, te3);
                trc = fminf(fmaxf(trc, -3.0f + 1e-6f), 3.0f - 1e-6f);
                float ca = fminf(fmaxf((trc - 1.0f) * 0.5f, -1.0f + 1e-7f), 1.0f - 1e-7f);
                accAng += acosf(ca) * 57.29577951308232f;

                // ortho: ||A^T A - I||_F^2 on clipped pred (pre-orthonorm.)
                float d11 = dot3(pa1, pa1), d22 = dot3(pa2, pa2), d33 = dot3(pa3, pa3);
                float d12 = dot3(pa1, pa2), d13 = dot3(pa1, pa3), d23 = dot3(pa2, pa3);
                accOrt += (d11 - 1.f) * (d11 - 1.f) + (d22 - 1.f) * (d22 - 1.f) +
                          (d33 - 1.f) * (d33 - 1.f) +
                          2.0f * (d12 * d12 + d13 * d13 + d23 * d23);

                // l2 on clipped pred = sum of squared elements = tr(A^T A)
                accL2 += d11 + d22 + d33;

                // fallback: (clipped pred - raw target)^2
#pragma unroll
                for (int i = 0; i < 9; ++i) {
                    float df = cp[i] - tg[i];
                    accFb += df * df;
                }
            }
        }
        __syncthreads();                     // done reading buf[cur]
        cur ^= 1;
        t = tn;
    }

    // wave32 reduction, then combine 8 waves through LDS
    float v[5] = {accCho, accAng, accOrt, accL2, accFb};
#pragma unroll
    for (int j = 0; j < 5; ++j) {
        float x = v[j];
        for (int off = 16; off > 0; off >>= 1) x += __shfl_xor(x, off, 32);
        v[j] = x;
    }
    int lane = (int)threadIdx.x & 31, wav = (int)threadIdx.x >> 5;
    if (lane == 0) {
#pragma unroll
        for (int j = 0; j < 5; ++j) red[wav * 5 + j] = v[j];
    }
    __syncthreads();
    if (threadIdx.x < 5) {
        float s = 0.f;
#pragma unroll
        for (int w = 0; w < 8; ++w) s += red[w * 5 + threadIdx.x];
        partials[blockIdx.x * 5 + threadIdx.x] = s;
    }
}

__global__ __launch_bounds__(256) void rot_loss_finalize(
    const float* __restrict__ partials, int nParts, long long nRows,
    float* __restrict__ out)
{
    __shared__ double sh[5 * 256];
    double a[5] = {0, 0, 0, 0, 0};
    for (int b = (int)threadIdx.x; b < nParts; b += 256) {
#pragma unroll
        for (int j = 0; j < 5; ++j) a[j] += (double)partials[b * 5 + j];
    }
#pragma unroll
    for (int j = 0; j < 5; ++j) sh[j * 256 + threadIdx.x] = a[j];
    __syncthreads();
    for (int s = 128; s > 0; s >>= 1) {
        if ((int)threadIdx.x < s) {
#pragma unroll
            for (int j = 0; j < 5; ++j)
                sh[j * 256 + threadIdx.x] += sh[j * 256 + threadIdx.x + s];
        }
        __syncthreads();
    }
    if (threadIdx.x == 0) {
        double B   = (double)nRows;
        double cho = sh[0], ang = sh[256], ort = sh[512], l2 = sh[768], fb = sh[1024];
        double total = cho / B + 0.1 * (ang / B) + 0.01 * (ort / B)
                     + 1e-4 * (l2 / (9.0 * B));
        double fbm = fb / (9.0 * B);
        out[0] = isnan(total) ? (float)fbm : (float)total;
    }
}

extern "C" void kernel_launch(void* const* d_in, const int* in_sizes, int n_in,
                              void* d_out, int out_size, void* d_ws, size_t ws_size,
                              hipStream_t stream)
{
    (void)n_in; (void)out_size; (void)ws_size;
    const float* pred = (const float*)d_in[0];
    const float* targ = (const float*)d_in[1];
    long long B = (long long)in_sizes[0] / 9;
    int nTiles  = (int)((B + TILE_ROWS - 1) / TILE_ROWS);
    int blocks  = nTiles < 4096 ? nTiles : 4096;
    float* partials = (float*)d_ws;

    rot_loss_main<<<blocks, 256, 0, stream>>>(pred, targ, partials, nTiles, B);
    rot_loss_finalize<<<1, 256, 0, stream>>>(partials, blocks, B, (float*)d_out);
}